// VQLatentSpace1D_23691039605499
// MI455X (gfx1250) — compile-verified
//
#include <hip/hip_runtime.h>
#include <math.h>

#define B_DIM 32
#define C_DIM 64
#define W_DIM 4096
#define KCODES 1024
#define N_ROWS (B_DIM * W_DIM)            /* 131072 */
#define TILE_W 128
#define Q_SIZE (B_DIM * C_DIM * W_DIM)    /* 8388608 */
#define ENC_OFF (Q_SIZE + 2)
#define ENC_SIZE ((size_t)N_ROWS * KCODES) /* 134217728 */
#define LDS_STRIDE 68                      /* pad: bank-conflict-free, keeps 8/16B align */

typedef __attribute__((ext_vector_type(2))) float v2f;
typedef __attribute__((ext_vector_type(4))) float v4f;
typedef __attribute__((ext_vector_type(8))) float v8f;
typedef __attribute__((ext_vector_type(4))) int   v4i;

typedef __attribute__((address_space(1))) v4i gv4i;
typedef __attribute__((address_space(3))) v4i lv4i;

// Async DMA: 16 bytes per lane, global -> LDS, tracked by ASYNCcnt.
__device__ __forceinline__ void async_copy16(const float* g, float* l) {
    __builtin_amdgcn_global_load_async_to_lds_b128(
        (gv4i*)g, (lv4i*)l, 0, 0);
}

// ---------------------------------------------------------------- init
__global__ void vq_init(float* __restrict__ lossSum, int* __restrict__ count) {
    int t = blockIdx.x * blockDim.x + threadIdx.x;
    if (t == 0) *lossSum = 0.0f;
    if (t < KCODES) count[t] = 0;
}

// ---------------------------------------------------------------- ||e_k||^2
__global__ void vq_enorm(const float* __restrict__ emb, float* __restrict__ eNorm) {
    int k = blockIdx.x * blockDim.x + threadIdx.x;
    if (k >= KCODES) return;
    const v4f* row = (const v4f*)(emb + (size_t)k * C_DIM);
    float s = 0.0f;
#pragma unroll
    for (int i = 0; i < 16; ++i) {
        v4f v = row[i];
        s += v.x * v.x + v.y * v.y + v.z * v.z + v.w * v.w;
    }
    eNorm[k] = s;
}

// ---------------------------------------------------------------- main
__global__ __launch_bounds__(256) void vq_main(
        const float* __restrict__ x, const float* __restrict__ emb,
        const float* __restrict__ eNorm, float* __restrict__ quant,
        float* __restrict__ enc, float* __restrict__ lossSum,
        int* __restrict__ count) {
    __shared__ float xT[TILE_W * LDS_STRIDE];        /* transposed input tile [w][c] */
    __shared__ float etile[2][16 * LDS_STRIDE];      /* double-buffered codebook tile */
    __shared__ int   idxLds[TILE_W];
    __shared__ float red[8];

    const int tid = threadIdx.x;
    const int n_base = blockIdx.x * TILE_W;
    const int b = n_base / W_DIM;
    const int w_base = n_base % W_DIM;

    // Phase 1: coalesced (over w) load of strided input, transpose into LDS.
    // Input is read exactly once -> non-temporal (don't churn L2).
#pragma unroll
    for (int i = 0; i < 32; ++i) {
        int lin = i * 256 + tid;
        int c = lin >> 7;        /* 0..63  */
        int w = lin & 127;       /* 0..127 */
        xT[w * LDS_STRIDE + c] = __builtin_nontemporal_load(
            &x[(size_t)b * (C_DIM * W_DIM) + (size_t)c * W_DIM + (w_base + w)]);
    }

    // Each thread's slot in the codebook staging (16B per lane, 4KB per tile)
    const int scode = tid >> 4;          /* 0..15 code within tile */
    const int scq   = (tid & 15) << 2;   /* 0..60 dim quad         */

    // Prologue: kick off async DMA of tile 0 while xT transpose settles
    async_copy16(&emb[(size_t)scode * C_DIM + scq],
                 &etile[0][scode * LDS_STRIDE + scq]);

    __syncthreads();

    const int wave = tid >> 5;
    const int lane = tid & 31;
    const int hlf  = lane >> 4;          /* lane half: 0 or 1 */
    const int l16  = lane & 15;
    const int mloc = wave * 16 + l16;    /* this lane's A-matrix row */

    // A fragments for all 16 k-steps (ISA: A 16x4 f32, lane=M, elems K={2h,2h+1}+4s)
    v2f a[16];
#pragma unroll
    for (int s = 0; s < 16; ++s)
        a[s] = *(const v2f*)&xT[mloc * LDS_STRIDE + 4 * s + 2 * hlf];

    float minV[8];
    int   minI[8];
#pragma unroll
    for (int j = 0; j < 8; ++j) { minV[j] = __builtin_inff(); minI[j] = 0; }

    // Phase 2: sweep codebook in 16-code tiles; async DMA of tile kc+1
    // overlaps the 16 WMMAs of tile kc (double buffer, one barrier/iter).
    for (int kc = 0; kc < KCODES / 16; ++kc) {
        const int cur = kc & 1;
        __builtin_amdgcn_s_wait_asynccnt(0);  /* own stage of tile kc done   */
        __syncthreads();                      /* everyone's stage kc visible;
                                                 everyone done reading buf cur^1 */
        if (kc + 1 < KCODES / 16)
            async_copy16(&emb[(size_t)((kc + 1) * 16 + scode) * C_DIM + scq],
                         &etile[cur ^ 1][scode * LDS_STRIDE + scq]);

        v8f acc = {0.f, 0.f, 0.f, 0.f, 0.f, 0.f, 0.f, 0.f};
#pragma unroll
        for (int s = 0; s < 16; ++s) {
            v2f bfrag = *(const v2f*)&etile[cur][l16 * LDS_STRIDE + 4 * s + 2 * hlf];
            acc = __builtin_amdgcn_wmma_f32_16x16x4_f32(
                false, a[s], false, bfrag, (short)0, acc, false, false);
        }

        int code = kc * 16 + l16;
        float en = eNorm[code];
#pragma unroll
        for (int j = 0; j < 8; ++j) {   /* C layout: M = j + 8*hlf, N = l16 */
            float sc = en - 2.0f * acc[j];
            if (sc < minV[j]) { minV[j] = sc; minI[j] = code; }
        }
    }

    // Cross-lane argmin reduction within each 16-lane group (lowest-index tie-break)
#pragma unroll
    for (int j = 0; j < 8; ++j) {
        float v = minV[j];
        int  ii = minI[j];
#pragma unroll
        for (int off = 1; off < 16; off <<= 1) {
            float ov = __shfl_xor(v, off, 32);
            int   oi = __shfl_xor(ii, off, 32);
            if (ov < v || (ov == v && oi < ii)) { v = ov; ii = oi; }
        }
        if (l16 == 0) idxLds[wave * 16 + hlf * 8 + j] = ii;
    }
    __syncthreads();

    // Phase 3a: quantized output (== straight-through value) + loss partial.
    // Streaming output -> non-temporal stores.
    float lsum = 0.0f;
#pragma unroll
    for (int i = 0; i < 32; ++i) {
        int lin = i * 256 + tid;
        int c = lin >> 7;
        int w = lin & 127;
        float q  = emb[(size_t)idxLds[w] * C_DIM + c];
        float xv = xT[w * LDS_STRIDE + c];
        __builtin_nontemporal_store(
            q, &quant[(size_t)b * (C_DIM * W_DIM) + (size_t)c * W_DIM + (w_base + w)]);
        float d = q - xv;
        lsum += d * d;
    }

    // Phase 3b: fused one-hot materialization — write all 128x1024 values
    // (1.0 at idx, 0.0 elsewhere). Every byte written exactly once, no
    // zero-fill pass, no scatter RMW. NT streaming stores (512 MB total).
    for (int i = tid; i < TILE_W * (KCODES / 2); i += 256) {
        int w  = i >> 9;                 /* row within tile   */
        int c2 = (i & 511) << 1;         /* even column index */
        int k  = idxLds[w];
        v2f v;
        v.x = (c2     == k) ? 1.0f : 0.0f;
        v.y = (c2 + 1 == k) ? 1.0f : 0.0f;
        __builtin_nontemporal_store(
            v, (v2f*)&enc[(size_t)(n_base + w) * KCODES + c2]);
    }

    // Histogram for perplexity
    if (tid < TILE_W) atomicAdd(&count[idxLds[tid]], 1);

    // Block loss reduction -> global atomic
#pragma unroll
    for (int off = 16; off > 0; off >>= 1) lsum += __shfl_down(lsum, off, 32);
    if (lane == 0) red[wave] = lsum;
    __syncthreads();
    if (tid == 0) {
        float s = 0.0f;
#pragma unroll
        for (int i = 0; i < 8; ++i) s += red[i];
        atomicAdd(lossSum, s);
    }
}

// ---------------------------------------------------------------- finalize
__global__ void vq_finalize(const float* __restrict__ lossSum,
                            const int* __restrict__ count,
                            float* __restrict__ outLoss,
                            float* __restrict__ outPerp) {
    __shared__ float red[32];
    int t = threadIdx.x;  /* 1024 threads, one per code */
    float p = (float)count[t] * (1.0f / (float)N_ROWS);
    float term = -p * logf(p + 1e-10f);
#pragma unroll
    for (int off = 16; off > 0; off >>= 1) term += __shfl_down(term, off, 32);
    if ((t & 31) == 0) red[t >> 5] = term;
    __syncthreads();
    if (t < 32) {
        float v = red[t];
#pragma unroll
        for (int off = 16; off > 0; off >>= 1) v += __shfl_down(v, off, 32);
        if (t == 0) {
            *outPerp = expf(v);
            *outLoss = 1.25f * (*lossSum) * (1.0f / (float)Q_SIZE);
        }
    }
}

// ---------------------------------------------------------------- launch
extern "C" void kernel_launch(void* const* d_in, const int* in_sizes, int n_in,
                              void* d_out, int out_size, void* d_ws, size_t ws_size,
                              hipStream_t stream) {
    (void)in_sizes; (void)n_in; (void)out_size; (void)ws_size;
    const float* x   = (const float*)d_in[0];   /* (32,64,4096) f32 */
    const float* emb = (const float*)d_in[1];   /* (1024,64)   f32 */

    float* out     = (float*)d_out;
    float* quant   = out;                 /* 8388608 */
    float* lossOut = out + Q_SIZE;        /* 1 */
    float* perpOut = out + Q_SIZE + 1;    /* 1 */
    float* enc     = out + ENC_OFF;       /* 134217728 */

    float* eNorm   = (float*)d_ws;                      /* 1024 f32 */
    int*   count   = (int*)((char*)d_ws + 4096);        /* 1024 i32 */
    float* lossSum = (float*)((char*)d_ws + 8192);      /* 1 f32    */

    vq_init    <<<1, 1024, 0, stream>>>(lossSum, count);
    vq_enorm   <<<4, 256, 0, stream>>>(emb, eNorm);
    vq_main    <<<N_ROWS / TILE_W, 256, 0, stream>>>(x, emb, eNorm, quant, enc,
                                                     lossSum, count);
    vq_finalize<<<1, 1024, 0, stream>>>(lossSum, count, lossOut, perpOut);
}